// GAT_HSPA_Cascade_76879914598934
// MI455X (gfx1250) — compile-verified
//
#include <hip/hip_runtime.h>
#include <hip/hip_bf16.h>

typedef float v2f __attribute__((ext_vector_type(2)));
typedef float v8f __attribute__((ext_vector_type(8)));

#define BN_EPS 1e-5f

// ---------------------------------------------------------------------------
// Tiled WMMA FP32 GEMM:  D[b] = epi( op(A[b]) * op(B[b]) )
//   op(A): MxK (TA!=0 -> A stored KxM), op(B): KxN (TB!=0 -> B stored NxK)
// 128 threads = 4 waves; block tile 64x128; each wave 32x64 = 2x4 WMMA frags.
// K staged through LDS in chunks of 16 (4 x k=4 WMMA steps).
// LDS tiles are k-pair interleaved v2f ([kpair][col]) so every A/B fragment
// is a single aligned ds_load_b64 straight into an even register pair
// (matches V_WMMA_F32_16X16X4_F32 operand layout: lane = k_hi*16 + m|n,
//  vgpr j <-> K = 2*kpair + j).  Row stride 80/144 v2f => the two lane
// halves hit disjoint 32-bank ranges.
// Epilogue fuses bias, folded BatchNorm, ReLU and residual add.
// ---------------------------------------------------------------------------
__global__ __launch_bounds__(128) void gemm_wmma_f32(
    const float* __restrict__ A, const float* __restrict__ B,
    float* __restrict__ D,
    int M, int N, int K, int lda, int ldb, int ldd,
    long long sA, long long sB, long long sD,
    int TA, int TB,
    const float* __restrict__ bias,
    const float* __restrict__ gamma, const float* __restrict__ beta,
    const float* __restrict__ mean, const float* __restrict__ var,
    int relu,
    const float* __restrict__ add, long long sAdd)
{
  __shared__ v2f As2[8 * 80];    // [kpair][m 0..63]  (+16 v2f pad)  5.0 KB
  __shared__ v2f Bs2[8 * 144];   // [kpair][n 0..127] (+16 v2f pad)  9.0 KB

  const int bz = blockIdx.z;
  const float* Ab = A + (size_t)bz * sA;
  const float* Bb = B + (size_t)bz * sB;
  float* Db = D + (size_t)bz * sD;

  const int Mb = blockIdx.y * 64;
  const int Nb = blockIdx.x * 128;

  const int tid  = threadIdx.x;
  const int lane = tid & 31;
  const int ln15 = lane & 15;
  const int kh   = lane >> 4;        // k_hi half-select per ISA 7.12.2
  const int wave = tid >> 5;
  const int mBase = (wave >> 1) * 32;
  const int nBase = (wave & 1) * 64;

  v8f c[2][4] = {};

  for (int k0 = 0; k0 < K; k0 += 16) {
    // cooperative global->LDS tile loads (zero-padded at edges)
    for (int e = tid; e < 16 * 64; e += 128) {     // A tile: 8 iters
      const int k = e >> 6, m = e & 63;
      const int gk = k0 + k;
      const int gm = Mb + m;
      float va = 0.0f;
      if (gk < K && gm < M)
        va = TA ? Ab[(size_t)gk * lda + gm] : Ab[(size_t)gm * lda + gk];
      ((float*)As2)[(((k >> 1) * 80) + m) * 2 + (k & 1)] = va;
    }
    for (int e = tid; e < 16 * 128; e += 128) {    // B tile: 16 iters
      const int k = e >> 7, n = e & 127;
      const int gk = k0 + k;
      const int gn = Nb + n;
      float vb = 0.0f;
      if (gk < K && gn < N)
        vb = TB ? Bb[(size_t)gn * ldb + gk] : Bb[(size_t)gk * ldb + gn];
      ((float*)Bs2)[(((k >> 1) * 144) + n) * 2 + (k & 1)] = vb;
    }
    __syncthreads();

    // speculative prefetch of next K-chunk (global_prefetch_b8)
    if (k0 + 16 < K) {
      const int gk = k0 + 16 + (tid >> 3);
      if (gk < K) {
        const int ga = (tid & 7) * 8;
        const int gb = (tid & 7) * 16;
        if (Mb + ga < M)
          __builtin_prefetch(TA ? &Ab[(size_t)gk * lda + Mb + ga]
                                : &Ab[(size_t)(Mb + ga) * lda + gk], 0, 1);
        if (Nb + gb < N)
          __builtin_prefetch(TB ? &Bb[(size_t)(Nb + gb) * ldb + gk]
                                : &Bb[(size_t)gk * ldb + Nb + gb], 0, 1);
      }
    }

    #pragma unroll
    for (int ks = 0; ks < 16; ks += 4) {
      const int p = (ks >> 1) + kh;              // k-pair row for this half
      v2f a[2], b[4];
      #pragma unroll
      for (int mt = 0; mt < 2; ++mt)
        a[mt] = As2[p * 80 + mBase + mt * 16 + ln15];
      #pragma unroll
      for (int nt = 0; nt < 4; ++nt)
        b[nt] = Bs2[p * 144 + nBase + nt * 16 + ln15];
      #pragma unroll
      for (int mt = 0; mt < 2; ++mt)
        #pragma unroll
        for (int nt = 0; nt < 4; ++nt)
          c[mt][nt] = __builtin_amdgcn_wmma_f32_16x16x4_f32(
              false, a[mt], false, b[nt], (short)0, c[mt][nt], false, false);
    }
    __syncthreads();
  }

  // fused epilogue + store (C/D layout: m = v + 8*k_hi, n = lane&15)
  #pragma unroll
  for (int mt = 0; mt < 2; ++mt) {
    #pragma unroll
    for (int nt = 0; nt < 4; ++nt) {
      const int gn = Nb + nBase + nt * 16 + ln15;
      #pragma unroll
      for (int v = 0; v < 8; ++v) {
        const int gm = Mb + mBase + mt * 16 + kh * 8 + v;
        if (gm < M && gn < N) {
          float val = c[mt][nt][v];
          if (bias) val += bias[gm];
          if (gamma) {
            const float s = gamma[gm] * rsqrtf(var[gm] + BN_EPS);
            val = val * s + (beta[gm] - mean[gm] * s);
          }
          if (relu) val = fmaxf(val, 0.0f);
          if (add) val += add[(size_t)bz * sAdd + (size_t)gm * ldd + gn];
          Db[(size_t)gm * ldd + gn] = val;
        }
      }
    }
  }
}

// ---------------------------------------------------------------------------
// Row softmax, width 49 (one thread per row; rows are tiny & L2-resident)
// ---------------------------------------------------------------------------
__global__ void softmax_rows(float* __restrict__ S, int rows, int w) {
  const int r = blockIdx.x * blockDim.x + threadIdx.x;
  if (r >= rows) return;
  float* p = S + (size_t)r * w;
  float m = -__builtin_inff();
  for (int i = 0; i < w; ++i) m = fmaxf(m, p[i]);
  float s = 0.0f;
  for (int i = 0; i < w; ++i) s += __expf(p[i] - m);
  const float inv = 1.0f / s;
  for (int i = 0; i < w; ++i) p[i] = __expf(p[i] - m) * inv;
}

// ---------------------------------------------------------------------------
// 3x3 SAME conv (Ci=256 -> Co=64) + folded BN + PReLU.
// Block 32x32 threads handles one (batch, 4-output-channel group):
// per-Ci 33x33 halo tile + all 4x256x9 weights cached in LDS (~41 KB).
// ---------------------------------------------------------------------------
__global__ __launch_bounds__(1024) void conv3x3_bn_prelu(
    const float* __restrict__ X, const float* __restrict__ W,
    const float* __restrict__ bias,
    const float* __restrict__ gamma, const float* __restrict__ beta,
    const float* __restrict__ mean, const float* __restrict__ var,
    const float* __restrict__ prelu_a, float* __restrict__ Y)
{
  const int b   = blockIdx.y;
  const int cog = blockIdx.x;            // 16 groups of 4 output channels
  const int tx = threadIdx.x, ty = threadIdx.y;
  const int tid = ty * 32 + tx;
  __shared__ float s_w[4 * 256 * 9];
  __shared__ float s_t[33 * 33];
  for (int i = tid; i < 4 * 256 * 9; i += 1024)
    s_w[i] = W[(size_t)cog * 4 * 256 * 9 + i];
  float acc[4] = {0.0f, 0.0f, 0.0f, 0.0f};
  const float* Xb = X + (size_t)b * 256 * 961;
  for (int ci = 0; ci < 256; ++ci) {
    __syncthreads();
    for (int i = tid; i < 33 * 33; i += 1024) {
      const int yy = i / 33 - 1, xx = i % 33 - 1;
      s_t[i] = (yy >= 0 && yy < 31 && xx >= 0 && xx < 31)
                   ? Xb[ci * 961 + yy * 31 + xx] : 0.0f;
    }
    __syncthreads();
    if (ty < 31 && tx < 31) {
      float v[9];
      #pragma unroll
      for (int kyy = 0; kyy < 3; ++kyy)
        #pragma unroll
        for (int kxx = 0; kxx < 3; ++kxx)
          v[kyy * 3 + kxx] = s_t[(ty + kyy) * 33 + tx + kxx];
      #pragma unroll
      for (int j = 0; j < 4; ++j) {
        const float* w = &s_w[(j * 256 + ci) * 9];
        float a = 0.0f;
        #pragma unroll
        for (int q = 0; q < 9; ++q) a += v[q] * w[q];
        acc[j] += a;
      }
    }
  }
  if (ty < 31 && tx < 31) {
    const float pa = prelu_a[0];
    #pragma unroll
    for (int j = 0; j < 4; ++j) {
      const int co = cog * 4 + j;
      const float s = gamma[co] * rsqrtf(var[co] + BN_EPS);
      float val = (acc[j] + bias[co]) * s + (beta[co] - mean[co] * s);
      val = val >= 0.0f ? val : pa * val;
      Y[((size_t)b * 64 + co) * 961 + ty * 31 + tx] = val;
    }
  }
}

// ---------------------------------------------------------------------------
// Sparsemax over rows of width 961 (in place). One 256-thread block per row:
// bitonic sort of 1024 (-inf padded) in LDS, Hillis-Steele inclusive scan,
// support size via shared atomic, thresholded output.
// ---------------------------------------------------------------------------
#define SMAX_N 961
#define SMAX_P 1024

__global__ __launch_bounds__(256) void sparsemax_rows(float* __restrict__ S) {
  const int row = blockIdx.x;
  float* rp = S + (size_t)row * SMAX_N;
  __shared__ float sv[SMAX_P];   // shifted values, original order (for output)
  __shared__ float sa[SMAX_P];   // sort buffer / scan ping
  __shared__ float sb[SMAX_P];   // descending sorted (kept)
  __shared__ float ss[SMAX_P];   // scan pong
  __shared__ float s_red[256];
  __shared__ int s_k;
  __shared__ float s_tau;
  const int t = threadIdx.x;

  for (int i = t; i < SMAX_P; i += 256)
    sv[i] = (i < SMAX_N) ? rp[i] : -__builtin_inff();
  __syncthreads();

  // row max
  float m = -__builtin_inff();
  for (int i = t; i < SMAX_P; i += 256) m = fmaxf(m, sv[i]);
  s_red[t] = m;
  __syncthreads();
  for (int off = 128; off > 0; off >>= 1) {
    if (t < off) s_red[t] = fmaxf(s_red[t], s_red[t + off]);
    __syncthreads();
  }
  const float rowmax = s_red[0];
  __syncthreads();

  for (int i = t; i < SMAX_P; i += 256) {
    const float v = (i < SMAX_N) ? sv[i] - rowmax : -__builtin_inff();
    sv[i] = v;
    sa[i] = v;
  }
  __syncthreads();

  // bitonic ascending sort of sa[0..1023]
  for (unsigned k = 2; k <= SMAX_P; k <<= 1) {
    for (unsigned j = k >> 1; j > 0; j >>= 1) {
      for (int i = t; i < SMAX_P; i += 256) {
        const unsigned ixj = (unsigned)i ^ j;
        if (ixj > (unsigned)i) {
          const float x = sa[i], y = sa[ixj];
          const bool up = (((unsigned)i & k) == 0);
          if ((x > y) == up) { sa[i] = y; sa[ixj] = x; }
        }
      }
      __syncthreads();
    }
  }

  // descending view -> sb (kept as srt)
  for (int i = t; i < SMAX_P; i += 256) sb[i] = sa[SMAX_P - 1 - i];
  __syncthreads();

  // inclusive scan of sb, ping-pong ss/sa
  float* src = sb;
  float* dst = ss;
  for (int off = 1; off < SMAX_P; off <<= 1) {
    for (int i = t; i < SMAX_P; i += 256)
      dst[i] = src[i] + ((i >= off) ? src[i - off] : 0.0f);
    __syncthreads();
    src = dst;
    dst = (dst == ss) ? sa : ss;
  }
  const float* cum = src;

  // support size k: count of ( (i+1)*srt[i] > cum[i]-1 )
  if (t == 0) s_k = 0;
  __syncthreads();
  int local = 0;
  for (int i = t; i < SMAX_P; i += 256)
    if ((float)(i + 1) * sb[i] > cum[i] - 1.0f) ++local;
  atomicAdd(&s_k, local);
  __syncthreads();
  if (t == 0) {
    const int k = s_k;
    s_tau = (cum[k - 1] - 1.0f) / (float)k;
  }
  __syncthreads();
  const float tau = s_tau;
  for (int i = t; i < SMAX_N; i += 256)
    rp[i] = fmaxf(sv[i] - tau, 0.0f);
}

// ---------------------------------------------------------------------------
// Host orchestration
// ---------------------------------------------------------------------------
extern "C" void kernel_launch(void* const* d_in, const int* in_sizes, int n_in,
                              void* d_out, int out_size, void* d_ws, size_t ws_size,
                              hipStream_t stream) {
  (void)in_sizes; (void)n_in; (void)out_size; (void)ws_size;
  const float* zf       = (const float*)d_in[0];
  const float* xf       = (const float*)d_in[1];
  const float* Wq       = (const float*)d_in[2];
  const float* bq       = (const float*)d_in[3];
  const float* Ws_      = (const float*)d_in[4];
  const float* bs       = (const float*)d_in[5];
  const float* Wg       = (const float*)d_in[6];
  const float* bg       = (const float*)d_in[7];
  const float* g_gamma  = (const float*)d_in[8];
  const float* g_beta   = (const float*)d_in[9];
  const float* g_mean   = (const float*)d_in[10];
  const float* g_var    = (const float*)d_in[11];
  const float* Wfi      = (const float*)d_in[12];
  const float* bfi      = (const float*)d_in[13];
  const float* fi_gamma = (const float*)d_in[14];
  const float* fi_beta  = (const float*)d_in[15];
  const float* fi_mean  = (const float*)d_in[16];
  const float* fi_var   = (const float*)d_in[17];
  const float* Wm       = (const float*)d_in[18];
  const float* bm       = (const float*)d_in[19];
  const float* m_gamma  = (const float*)d_in[20];
  const float* m_beta   = (const float*)d_in[21];
  const float* m_mean   = (const float*)d_in[22];
  const float* m_var    = (const float*)d_in[23];
  const float* prelu_a  = (const float*)d_in[24];

  float* ws = (float*)d_ws;
  const int B = 32, C = 256, NX = 961, NZ = 49, CM = 64;

  // Lifetime-aliased workspace layout (floats). Early-stage buffers live
  // inside the region later reused for the big S/P matrix.
  const size_t off_xt  = 0;                                   // [B][C][NX]
  const size_t off_cat = off_xt + (size_t)B * C * NX;         // [B][2C][NX]
  const size_t off_zt  = off_cat + (size_t)B * 2 * C * NX;    // [B][C][NZ]
  const size_t off_zg  = off_zt + (size_t)B * C * NZ;         // [B][C][NZ]
  const size_t off_sim = off_zg + (size_t)B * C * NZ;         // [B][NX][NZ]
  const size_t off_P   = 0;                                   // [B][NX][NX]
  const size_t off_gat = off_P + (size_t)B * NX * NX;         // [B][C][NX]
  const size_t off_e   = off_gat + (size_t)B * C * NX;        // [B][CM][NX]

  auto gemm = [&](const float* A, const float* Bm, float* D,
                  int M, int N, int K, int lda, int ldb, int ldd,
                  long long sA, long long sB, long long sD, int TA, int TB,
                  const float* bias, const float* gamma, const float* beta,
                  const float* mean, const float* var, int relu,
                  const float* add, long long sAdd) {
    dim3 g((unsigned)((N + 127) / 128), (unsigned)((M + 63) / 64), (unsigned)B);
    gemm_wmma_f32<<<g, 128, 0, stream>>>(A, Bm, D, M, N, K, lda, ldb, ldd,
                                         sA, sB, sD, TA, TB, bias, gamma, beta,
                                         mean, var, relu, add, sAdd);
  };

  // 1) xf_t = Wq @ xf + bq                        [C x NX]
  gemm(Wq, xf, ws + off_xt, C, NX, C, C, NX, NX,
       0, (long long)C * NX, (long long)C * NX, 0, 0,
       bq, nullptr, nullptr, nullptr, nullptr, 0, nullptr, 0);
  // 2) zf_t = Ws @ zf + bs                        [C x NZ]
  gemm(Ws_, zf, ws + off_zt, C, NZ, C, C, NZ, NZ,
       0, (long long)C * NZ, (long long)C * NZ, 0, 0,
       bs, nullptr, nullptr, nullptr, nullptr, 0, nullptr, 0);
  // 3) xf_g = relu(bn(Wg @ xf + bg)) -> cat rows [C..2C)
  gemm(Wg, xf, ws + off_cat + (size_t)C * NX, C, NX, C, C, NX, NX,
       0, (long long)C * NX, (long long)2 * C * NX, 0, 0,
       bg, g_gamma, g_beta, g_mean, g_var, 1, nullptr, 0);
  // 4) zf_g = relu(bn(Wg @ zf + bg))              [C x NZ]
  gemm(Wg, zf, ws + off_zg, C, NZ, C, C, NZ, NZ,
       0, (long long)C * NZ, (long long)C * NZ, 0, 0,
       bg, g_gamma, g_beta, g_mean, g_var, 1, nullptr, 0);
  // 5) sim = xf_t^T @ zf_t                        [NX x NZ]
  gemm(ws + off_xt, ws + off_zt, ws + off_sim, NX, NZ, C, NX, NZ, NZ,
       (long long)C * NX, (long long)C * NZ, (long long)NX * NZ, 1, 0,
       nullptr, nullptr, nullptr, nullptr, nullptr, 0, nullptr, 0);
  // 6) softmax over NZ
  {
    const int rows = B * NX;
    softmax_rows<<<(rows + 255) / 256, 256, 0, stream>>>(ws + off_sim, rows, NZ);
  }
  // 7) emb = zf_g @ attn^T -> cat rows [0..C)
  gemm(ws + off_zg, ws + off_sim, ws + off_cat, C, NX, NZ, NZ, NZ, NX,
       (long long)C * NZ, (long long)NX * NZ, (long long)2 * C * NX, 0, 1,
       nullptr, nullptr, nullptr, nullptr, nullptr, 0, nullptr, 0);
  // 8) gat = relu(bn(Wfi @ cat + bfi))            [C x NX]
  gemm(Wfi, ws + off_cat, ws + off_gat, C, NX, 2 * C, 2 * C, NX, NX,
       0, (long long)2 * C * NX, (long long)C * NX, 0, 0,
       bfi, fi_gamma, fi_beta, fi_mean, fi_var, 1, nullptr, 0);
  // 9) e = prelu(bn(conv3x3(gat) + bm))           [CM x NX]
  conv3x3_bn_prelu<<<dim3(16, (unsigned)B), dim3(32, 32), 0, stream>>>(
      ws + off_gat, Wm, bm, m_gamma, m_beta, m_mean, m_var, prelu_a,
      ws + off_e);
  // 10) S = e^T @ e                               [NX x NX]
  gemm(ws + off_e, ws + off_e, ws + off_P, NX, NX, CM, NX, NX, NX,
       (long long)CM * NX, (long long)CM * NX, (long long)NX * NX, 1, 0,
       nullptr, nullptr, nullptr, nullptr, nullptr, 0, nullptr, 0);
  // 11) P = sparsemax_rows(S), in place
  sparsemax_rows<<<B * NX, 256, 0, stream>>>(ws + off_P);
  // 12) out = gat @ P + gat                       [C x NX]
  gemm(ws + off_gat, ws + off_P, (float*)d_out, C, NX, NX, NX, NX, NX,
       (long long)C * NX, (long long)NX * NX, (long long)C * NX, 0, 0,
       nullptr, nullptr, nullptr, nullptr, nullptr, 0,
       ws + off_gat, (long long)C * NX);
}